// Find_Ring_Bonds_64682207477991
// MI455X (gfx1250) — compile-verified
//
#include <hip/hip_runtime.h>
#include <hip/hip_bf16.h>
#include <stdint.h>

#define NATOMS 80
#define NDEG   4
#define NRINGS 10
#define RSIZE  8
#define BONDS  (NATOMS * NDEG)    // 320 bonds per molecule
#define RSLOTS (NRINGS * RSIZE)   // 80 ring slots per molecule
#define MPB    4                  // molecules per block
#define TPB    (MPB * NATOMS)     // 320 threads = 10 wave32 waves

// Block = dim3(80, 4): threadIdx.x = atom/slot id, threadIdx.y = molecule.
// 2D launch avoids the mul-based div/mod-by-80 chain in the preamble.
__global__ __launch_bounds__(TPB) void Find_Ring_Bonds_kernel(
    const float* __restrict__ edges,  // [B, 80, 4]  float, -1 = null neighbor
    const int*   __restrict__ rings,  // [B, 10, 8]  int,   -1 = padding
    float*       __restrict__ out)    // [B, 80, 4, 1]
{
    __shared__ int      rings_s[MPB * RSLOTS];   // async DMA target
    __shared__ uint32_t ringmask[MPB * NATOMS];  // bit r set => atom in ring r

    const int    t   = threadIdx.x;              // atom / ring-slot id (0..79)
    const int    m   = threadIdx.y;              // molecule within block (0..3)
    const size_t mol = (size_t)blockIdx.x * MPB + m;

    // Per-thread edge row as one b128 load (LOADcnt) -- overlaps with the
    // rings -> LDS async DMA below (ASYNCcnt): independent counters.
    const float4 ef4 = ((const float4*)edges)[mol * NATOMS + t];

    // CDNA5 async global->LDS DMA of this molecule's 320 B ring table:
    // 20 lanes x b128, no VGPR round-trip.
    if (t < (RSLOTS / 4)) {
        const int* gsrc    = rings + mol * RSLOTS + t * 4;
        uint32_t   lds_off = (uint32_t)(uintptr_t)&rings_s[m * RSLOTS + t * 4];
        asm volatile("global_load_async_to_lds_b128 %0, %1, off"
                     :: "v"(lds_off), "v"(gsrc)
                     : "memory");
    }
    ringmask[m * NATOMS + t] = 0u;               // every thread zeroes one mask entry

    // Drain this wave's async DMA, then block barrier so all waves see the
    // staged ring tables and zeroed masks.
    asm volatile("s_wait_asynccnt 0" ::: "memory");
    __syncthreads();

    // Scatter: ring slot t belongs to ring (t/8); set that ring's bit on its atom.
    {
        const int atom = rings_s[m * RSLOTS + t];
        if (atom >= 0 && atom < NATOMS) {
            atomicOr(&ringmask[m * NATOMS + atom], 1u << (t >> 3));
        }
    }
    __syncthreads();

    // Bond (t, d) is a ring bond iff the two endpoint masks share a ring bit.
    const uint32_t* mask = &ringmask[m * NATOMS];
    const uint32_t  ma   = mask[t];

    const int e0 = (int)ef4.x;
    const int e1 = (int)ef4.y;
    const int e2 = (int)ef4.z;
    const int e3 = (int)ef4.w;

    float4 o;
    o.x = (e0 >= 0 && e0 < NATOMS && (ma & mask[e0])) ? 1.0f : 0.0f;
    o.y = (e1 >= 0 && e1 < NATOMS && (ma & mask[e1])) ? 1.0f : 0.0f;
    o.z = (e2 >= 0 && e2 < NATOMS && (ma & mask[e2])) ? 1.0f : 0.0f;
    o.w = (e3 >= 0 && e3 < NATOMS && (ma & mask[e3])) ? 1.0f : 0.0f;

    ((float4*)out)[mol * NATOMS + t] = o;        // one b128 store per thread
}

extern "C" void kernel_launch(void* const* d_in, const int* in_sizes, int n_in,
                              void* d_out, int out_size, void* d_ws, size_t ws_size,
                              hipStream_t stream) {
    const float* edges = (const float*)d_in[0];  // [B, 80, 4] float32
    const int*   rings = (const int*)d_in[1];    // [B, 10, 8] int32
    float*       out   = (float*)d_out;          // [B, 80, 4, 1] float32

    const int B = in_sizes[0] / BONDS;           // 16384
    Find_Ring_Bonds_kernel<<<dim3(B / MPB), dim3(NATOMS, MPB), 0, stream>>>(edges, rings, out);
}